// EKTM_42855183679642
// MI455X (gfx1250) — compile-verified
//
#include <hip/hip_runtime.h>
#include <math.h>

#define SEQ_LEN 64
#define EMB 128
#define TH 128          // per-direction topic hidden
#define G3 384          // 3*TH
#define TOPICD 256
#define KCNT 16384
#define KE 64
#define SH 512
#define IN_S 513
#define GSZ 1536        // 3*SH

typedef __attribute__((ext_vector_type(2))) float v2f;
typedef __attribute__((ext_vector_type(8))) float v8f;

__device__ __forceinline__ float sigf(float x) { return 1.0f / (1.0f + expf(-x)); }

// ---------------- 1. topic bidirectional GRU (grid=2 dirs, block=384) -------
__global__ void topic_gru_kernel(const int* __restrict__ topic, const float* __restrict__ emb,
                                 const float* WihF, const float* WhhF, const float* bihF, const float* bhhF,
                                 const float* WihB, const float* WhhB, const float* bihB, const float* bhhB,
                                 float* __restrict__ ys) {
  __shared__ float xs[EMB], hs[TH], gi[G3], gh[G3];
  const int tid = threadIdx.x;
  const int dir = blockIdx.x;
  const float* Wih = dir ? WihB : WihF;
  const float* Whh = dir ? WhhB : WhhF;
  const float* bih = dir ? bihB : bihF;
  const float* bhh = dir ? bhhB : bhhF;
  if (tid < TH) hs[tid] = 0.0f;
  __syncthreads();
  for (int t = 0; t < SEQ_LEN; ++t) {
    const int tt = dir ? (SEQ_LEN - 1 - t) : t;
    if (tid < EMB) xs[tid] = emb[(long)topic[tt] * EMB + tid];
    __syncthreads();
    float ai = bih[tid], ah = bhh[tid];
    const float* wi = Wih + tid * EMB;
    const float* wh = Whh + tid * TH;
    #pragma unroll 8
    for (int c = 0; c < EMB; ++c) ai += wi[c] * xs[c];
    #pragma unroll 8
    for (int c = 0; c < TH; ++c)  ah += wh[c] * hs[c];
    gi[tid] = ai; gh[tid] = ah;
    __syncthreads();
    if (tid < TH) {
      float r = sigf(gi[tid] + gh[tid]);
      float z = sigf(gi[TH + tid] + gh[TH + tid]);
      float n = tanhf(gi[2 * TH + tid] + r * gh[2 * TH + tid]);
      float h2 = (1.0f - z) * n + z * hs[tid];
      hs[tid] = h2;
      ys[(dir * SEQ_LEN + tt) * TH + tid] = h2;
    }
    __syncthreads();
  }
}

// ---------------- 2. v = max over time of [yf|yb] (block=256) ---------------
__global__ void vmax_kernel(const float* __restrict__ ys, float* __restrict__ v) {
  const int j = threadIdx.x;           // 0..255
  const int dir = j / TH, jj = j % TH;
  float m = -INFINITY;
  for (int t = 0; t < SEQ_LEN; ++t) m = fmaxf(m, ys[(dir * SEQ_LEN + t) * TH + jj]);
  v[j] = m;
}

// ---------------- 3. kn = Wk @ knowledge + bk (grid=64, block=256) ----------
__global__ void kn_kernel(const float* __restrict__ Wk, const float* __restrict__ bk,
                          const float* __restrict__ know, float* __restrict__ kn) {
  __shared__ float red[256];
  const int row = blockIdx.x, tid = threadIdx.x;
  float acc = 0.0f;
  const float* wr = Wk + (long)row * KCNT;
  for (int c = tid; c < KCNT; c += 256) acc += wr[c] * know[c];
  red[tid] = acc; __syncthreads();
  for (int s = 128; s > 0; s >>= 1) { if (tid < s) red[tid] += red[tid + s]; __syncthreads(); }
  if (tid == 0) kn[row] = red[0] + bk[row];
}

// ---------------- 4. alpha = KM @ kn (grid=16384, block=64) -----------------
__global__ void alpha_kernel(const float* __restrict__ KM, const float* __restrict__ kn,
                             float* __restrict__ alpha) {
  __shared__ float red[64];
  const int row = blockIdx.x, tid = threadIdx.x;
  red[tid] = KM[(long)row * KE + tid] * kn[tid];
  __syncthreads();
  for (int s = 32; s > 0; s >>= 1) { if (tid < s) red[tid] += red[tid + s]; __syncthreads(); }
  if (tid == 0) alpha[row] = red[0];
}

// ---------------- 5. beta = softmax(alpha) (1 block of 1024) ----------------
__global__ void softmax_kernel(const float* __restrict__ alpha, float* __restrict__ beta) {
  __shared__ float red[1024];
  const int tid = threadIdx.x;
  float m = -INFINITY;
  for (int k = tid; k < KCNT; k += 1024) m = fmaxf(m, alpha[k]);
  red[tid] = m; __syncthreads();
  for (int s = 512; s > 0; s >>= 1) { if (tid < s) red[tid] = fmaxf(red[tid], red[tid + s]); __syncthreads(); }
  const float gmax = red[0]; __syncthreads();
  float sum = 0.0f;
  for (int k = tid; k < KCNT; k += 1024) sum += expf(alpha[k] - gmax);
  red[tid] = sum; __syncthreads();
  for (int s = 512; s > 0; s >>= 1) { if (tid < s) red[tid] += red[tid + s]; __syncthreads(); }
  const float inv = 1.0f / red[0];
  for (int k = tid; k < KCNT; k += 1024) beta[k] = expf(alpha[k] - gmax) * inv;
}

// ---------------- 6. hkp = beta @ h_initial (partials + reduce) -------------
__global__ void hkp_part_kernel(const float* __restrict__ beta, const float* __restrict__ h0,
                                float* __restrict__ part) {
  const int j = threadIdx.x, b = blockIdx.x;
  float acc = 0.0f;
  for (int r = 0; r < 128; ++r) {
    const int k = b * 128 + r;
    acc += beta[k] * h0[(long)k * SH + j];
  }
  part[b * SH + j] = acc;
}
__global__ void hkp_reduce_kernel(const float* __restrict__ part, float* __restrict__ hkp) {
  const int j = threadIdx.x;
  float acc = 0.0f;
  for (int b = 0; b < 128; ++b) acc += part[b * SH + j];
  hkp[j] = acc;
}

// ---------------- 7a. x_in = [v*pos, v*(1-pos), score] ----------------------
__global__ void xin_kernel(const float* __restrict__ v, const float* __restrict__ score,
                           float* __restrict__ xin) {
  const int i = threadIdx.x;
  const float s = score[0];
  const float pos = (s >= 0.5f) ? 1.0f : 0.0f;
  if (i < TOPICD)            xin[i] = v[i] * pos;
  else if (i < 2 * TOPICD)   xin[i] = v[i - TOPICD] * (1.0f - pos);
  else                       xin[i] = s;
}

// ---------------- 7b. g = W_ih_s @ x_in (rank-1 trick) (grid=1536) ----------
__global__ void g_kernel(const float* __restrict__ Wih_s, const float* __restrict__ xin,
                         float* __restrict__ g) {
  __shared__ float red[128];
  const int row = blockIdx.x, tid = threadIdx.x;
  float acc = 0.0f;
  const float* wr = Wih_s + (long)row * IN_S;
  for (int c = tid; c < IN_S; c += 128) acc += wr[c] * xin[c];
  red[tid] = acc; __syncthreads();
  for (int s = 64; s > 0; s >>= 1) { if (tid < s) red[tid] += red[tid + s]; __syncthreads(); }
  if (tid == 0) g[row] = red[0];
}

// ---------------- 8. predict_score = [v|hkp] . Ws + bs ----------------------
__global__ void predict_kernel(const float* __restrict__ v, const float* __restrict__ hkp,
                               const float* __restrict__ Ws, const float* __restrict__ bs,
                               float* __restrict__ out0) {
  __shared__ float red[256];
  const int tid = threadIdx.x;
  float acc = 0.0f;
  for (int i = tid; i < TOPICD + SH; i += 256) {
    const float pv = (i < TOPICD) ? v[i] : hkp[i - TOPICD];
    acc += pv * Ws[i];
  }
  red[tid] = acc; __syncthreads();
  for (int s = 128; s > 0; s >>= 1) { if (tid < s) red[tid] += red[tid + s]; __syncthreads(); }
  if (tid == 0) out0[0] = red[0] + bs[0];
}

// ---------------- 9. fused WMMA slot-GRU: gh GEMM + gates -------------------
// D = A x B + C, f32 WMMA 16x16 with K=4.
// Each wave: TWO 16-row A tiles (rows k0..k0+15 and k0+16..k0+31) against the
// SAME three B tiles (gates r,z,n at cols j, j+512, j+1024) -> 6 WMMA per
// 5 float2 loads, amortizing the W_hh_s stream across two row tiles.
// A layout : lane L holds A[L%16, 2*(L/16)+{0,1}] in a float2
// B layout : lane L holds B[2*(L/16)+{0,1}, L%16] = Whh[j0+L%16, c+2*(L/16)+{0,1}]
// C layout : element e of v8f on lane L = C[e + 8*(L/16), L%16]
__global__ void __launch_bounds__(256) slot_gru_kernel(
    const float* __restrict__ h0, const float* __restrict__ Whh,
    const float* __restrict__ bih, const float* __restrict__ bhh,
    const float* __restrict__ beta, const float* __restrict__ g,
    float* __restrict__ out) {
  const int lane  = threadIdx.x & 31;
  const int wave  = threadIdx.x >> 5;
  const int k0    = blockIdx.x * 32;                 // 32 slot rows per block
  const int j0    = blockIdx.y * 128 + wave * 16;    // 16 hidden cols per wave
  const int nrow  = lane & 15;                       // M (for A) / N (for B)
  const int khalf = (lane >> 4) << 1;                // 0 or 2

  const float* aptr0 = h0  + (long)(k0 + nrow) * SH + khalf;
  const float* aptr1 = aptr0 + 16L * SH;
  const float* brptr = Whh + (long)(j0 + nrow) * SH + khalf;
  const float* bzptr = Whh + (long)(j0 + SH + nrow) * SH + khalf;
  const float* bnptr = Whh + (long)(j0 + 2 * SH + nrow) * SH + khalf;

  v8f accr[2] = {}; v8f accz[2] = {}; v8f accn[2] = {};
  #pragma unroll 2
  for (int c = 0; c < SH; c += 4) {
    const v2f a0 = *(const v2f*)(aptr0 + c);
    const v2f a1 = *(const v2f*)(aptr1 + c);
    const v2f br = *(const v2f*)(brptr + c);
    const v2f bz = *(const v2f*)(bzptr + c);
    const v2f bn = *(const v2f*)(bnptr + c);
    accr[0] = __builtin_amdgcn_wmma_f32_16x16x4_f32(false, a0, false, br, (short)0, accr[0], false, false);
    accz[0] = __builtin_amdgcn_wmma_f32_16x16x4_f32(false, a0, false, bz, (short)0, accz[0], false, false);
    accn[0] = __builtin_amdgcn_wmma_f32_16x16x4_f32(false, a0, false, bn, (short)0, accn[0], false, false);
    accr[1] = __builtin_amdgcn_wmma_f32_16x16x4_f32(false, a1, false, br, (short)0, accr[1], false, false);
    accz[1] = __builtin_amdgcn_wmma_f32_16x16x4_f32(false, a1, false, bz, (short)0, accz[1], false, false);
    accn[1] = __builtin_amdgcn_wmma_f32_16x16x4_f32(false, a1, false, bn, (short)0, accn[1], false, false);
  }

  // Epilogue: gi[k,j] = beta[k]*g[j] + bih[j]; gh[k,j] = acc + bhh[j]
  const int j = j0 + nrow;
  const float gr = g[j],          gz = g[j + SH],      gn = g[j + 2 * SH];
  const float bir = bih[j],       biz = bih[j + SH],   bin_ = bih[j + 2 * SH];
  const float bhr = bhh[j],       bhz = bhh[j + SH],   bhn = bhh[j + 2 * SH];
  #pragma unroll
  for (int m = 0; m < 2; ++m) {
    const int kbase = k0 + m * 16 + ((lane >> 4) << 3);
    #pragma unroll
    for (int e = 0; e < 8; ++e) {
      const int k = kbase + e;
      const float bet = beta[k];
      const float hv  = h0[(long)k * SH + j];
      const float r = sigf(bet * gr + bir + accr[m][e] + bhr);
      const float z = sigf(bet * gz + biz + accz[m][e] + bhz);
      const float n = tanhf(bet * gn + bin_ + r * (accn[m][e] + bhn));
      out[(long)k * SH + j] = (1.0f - z) * n + z * hv;
    }
  }
}

extern "C" void kernel_launch(void* const* d_in, const int* in_sizes, int n_in,
                              void* d_out, int out_size, void* d_ws, size_t ws_size,
                              hipStream_t stream) {
  (void)in_sizes; (void)n_in; (void)out_size; (void)ws_size;
  const int*   topic = (const int*)d_in[0];
  const float* know  = (const float*)d_in[1];
  const float* score = (const float*)d_in[2];
  const float* emb   = (const float*)d_in[3];
  const float* WihF  = (const float*)d_in[4];
  const float* WhhF  = (const float*)d_in[5];
  const float* bihF  = (const float*)d_in[6];
  const float* bhhF  = (const float*)d_in[7];
  const float* WihB  = (const float*)d_in[8];
  const float* WhhB  = (const float*)d_in[9];
  const float* bihB  = (const float*)d_in[10];
  const float* bhhB  = (const float*)d_in[11];
  const float* Wk    = (const float*)d_in[12];
  const float* bk    = (const float*)d_in[13];
  const float* KM    = (const float*)d_in[14];
  const float* h0    = (const float*)d_in[15];
  const float* Ws    = (const float*)d_in[16];
  const float* bs    = (const float*)d_in[17];
  const float* WihS  = (const float*)d_in[18];
  const float* WhhS  = (const float*)d_in[19];
  const float* bihS  = (const float*)d_in[20];
  const float* bhhS  = (const float*)d_in[21];

  float* ws    = (float*)d_ws;
  float* ys    = ws;              // 2*64*128 = 16384
  float* v     = ws + 16384;      // 256
  float* kn    = ws + 16640;      // 64
  float* alpha = ws + 16704;      // 16384
  float* beta  = ws + 33088;      // 16384
  float* part  = ws + 49472;      // 128*512 = 65536
  float* hkp   = ws + 115008;     // 512
  float* xin   = ws + 115520;     // 513 (pad to 520)
  float* g     = ws + 116040;     // 1536  (total ~117576 floats ~ 470 KB)

  float* out   = (float*)d_out;   // [0] = predict_score, [1..] = h_new[16384*512]

  topic_gru_kernel<<<2, G3, 0, stream>>>(topic, emb, WihF, WhhF, bihF, bhhF,
                                         WihB, WhhB, bihB, bhhB, ys);
  vmax_kernel<<<1, TOPICD, 0, stream>>>(ys, v);
  kn_kernel<<<KE, 256, 0, stream>>>(Wk, bk, know, kn);
  alpha_kernel<<<KCNT, KE, 0, stream>>>(KM, kn, alpha);
  softmax_kernel<<<1, 1024, 0, stream>>>(alpha, beta);
  hkp_part_kernel<<<128, SH, 0, stream>>>(beta, h0, part);
  hkp_reduce_kernel<<<1, SH, 0, stream>>>(part, hkp);
  xin_kernel<<<1, IN_S, 0, stream>>>(v, score, xin);
  g_kernel<<<GSZ, 128, 0, stream>>>(WihS, xin, g);
  predict_kernel<<<1, 256, 0, stream>>>(v, hkp, Ws, bs, out);
  slot_gru_kernel<<<dim3(KCNT / 32, SH / 128), 256, 0, stream>>>(
      h0, WhhS, bihS, bhhS, beta, g, out + 1);
}